// KnotGAT_10007273800008
// MI455X (gfx1250) — compile-verified
//
#include <hip/hip_runtime.h>
#include <math.h>

#define HIDDEN 64
#define NEG_SLOPE 0.2f

typedef __attribute__((ext_vector_type(2))) float v2f;
typedef __attribute__((ext_vector_type(8))) float v8f;

// ---------------------------------------------------------------------------
// Order-preserving float -> uint encoding so segment-max uses the native
// integer atomic max (GLOBAL_ATOMIC_MAX_U32) instead of a CAS loop.
// ---------------------------------------------------------------------------
__device__ __forceinline__ unsigned enc_f32(float f) {
    unsigned u = __float_as_uint(f);
    return (u & 0x80000000u) ? ~u : (u | 0x80000000u);
}
__device__ __forceinline__ float dec_f32(unsigned u) {
    return (u & 0x80000000u) ? __uint_as_float(u ^ 0x80000000u)
                             : __uint_as_float(~u);
}

// ---------------------------------------------------------------------------
// h[i] = embed_table[x[i]]
// ---------------------------------------------------------------------------
__global__ void k_embed(const int* __restrict__ x, const float* __restrict__ table,
                        float* __restrict__ h, int N) {
    int idx = blockIdx.x * blockDim.x + threadIdx.x;
    if (idx >= N * HIDDEN) return;
    int node = idx >> 6, f = idx & 63;
    h[idx] = table[x[node] * HIDDEN + f];
}

// ---------------------------------------------------------------------------
// Per-layer re-init: out accumulator = 0, m = 0u (below encoded(-inf)), denom = 0
// ---------------------------------------------------------------------------
__global__ void k_layer_init(float* __restrict__ hout, unsigned* __restrict__ m_enc,
                             float* __restrict__ denom, int N) {
    int idx = blockIdx.x * blockDim.x + threadIdx.x;
    if (idx < N * HIDDEN) hout[idx] = 0.0f;
    if (idx < N) { m_enc[idx] = 0u; denom[idx] = 0.0f; }
}

// ---------------------------------------------------------------------------
// hW = h @ W  via V_WMMA_F32_16X16X4_F32.
// Block = 256 threads = 8 waves; wave w computes rows [blk*128 + w*16, +16)
// against all 64 columns (4 tiles of 16 cols), K = 64 in 16 steps of 4.
//
// f32 WMMA 16x16x4 VGPR layouts (ISA 7.12.2):
//   A (16x4):  lane<16 holds row=lane,   {K=kb, K=kb+1} with kb = 2*(lane>>4)
//   B (4x16):  lane<16 holds col=lane,   VGPR0={K=kb}, VGPR1={K=kb+1}
//   C/D:       VGPR v: lanes0-15 -> M=v, lanes16-31 -> M=v+8, N=lane&15
// ---------------------------------------------------------------------------
__global__ void k_gemm_hW(const float* __restrict__ h, const float* __restrict__ W,
                          float* __restrict__ hW, int N) {
    const int lane = threadIdx.x & 31;
    const int wave = threadIdx.x >> 5;
    const int row_base = blockIdx.x * 128 + wave * 16;
    if (row_base >= N) return;               // uniform per wave: EXEC stays all-1s

    const int half = lane >> 4;              // 0 or 1
    const int l16  = lane & 15;

    int arow = row_base + l16;               // clamp so loads stay in-bounds,
    if (arow >= N) arow = N - 1;             // EXEC stays full for WMMA

    v8f acc[4] = {};
    #pragma unroll
    for (int k = 0; k < HIDDEN; k += 4) {
        const int kb = k + half * 2;
        v2f a;
        a.x = h[arow * HIDDEN + kb];
        a.y = h[arow * HIDDEN + kb + 1];
        #pragma unroll
        for (int t = 0; t < 4; ++t) {
            const int col = t * 16 + l16;
            v2f b;
            b.x = W[kb * HIDDEN + col];
            b.y = W[(kb + 1) * HIDDEN + col];
            acc[t] = __builtin_amdgcn_wmma_f32_16x16x4_f32(
                false, a, false, b, (short)0, acc[t], false, false);
        }
    }

    #pragma unroll
    for (int t = 0; t < 4; ++t) {
        #pragma unroll
        for (int v = 0; v < 8; ++v) {
            const int r = row_base + v + 8 * half;
            if (r < N) hW[r * HIDDEN + t * 16 + l16] = acc[t][v];
        }
    }
}

// ---------------------------------------------------------------------------
// s_src[i] = hW[i] . a_src ; s_dst[i] = hW[i] . a_dst
// ---------------------------------------------------------------------------
__global__ void k_node_scores(const float* __restrict__ hW,
                              const float* __restrict__ a_s,
                              const float* __restrict__ a_d,
                              float* __restrict__ ssrc, float* __restrict__ sdst,
                              int N) {
    int i = blockIdx.x * blockDim.x + threadIdx.x;
    if (i >= N) return;
    const float4* row = (const float4*)(hW + (size_t)i * HIDDEN);
    const float4* as4 = (const float4*)a_s;
    const float4* ad4 = (const float4*)a_d;
    float s0 = 0.0f, s1 = 0.0f;
    #pragma unroll
    for (int c = 0; c < HIDDEN / 4; ++c) {
        float4 v = row[c], a = as4[c], b = ad4[c];
        s0 += v.x * a.x + v.y * a.y + v.z * a.z + v.w * a.w;
        s1 += v.x * b.x + v.y * b.y + v.z * b.z + v.w * b.w;
    }
    ssrc[i] = s0;
    sdst[i] = s1;
}

// ---------------------------------------------------------------------------
// Edge pass 1: e = leaky_relu(s_src[src] + s_dst[dst]); segment-max into m.
// Edges [0,E) come from edge_index; [E,Ep) are self loops.
// ---------------------------------------------------------------------------
__global__ void k_edge_score(const int* __restrict__ src, const int* __restrict__ dst,
                             const float* __restrict__ ssrc, const float* __restrict__ sdst,
                             float* __restrict__ e_buf, unsigned* __restrict__ m_enc,
                             int E, int Ep) {
    int e = blockIdx.x * blockDim.x + threadIdx.x;
    if (e >= Ep) return;
    int s, d;
    if (e < E) { s = src[e]; d = dst[e]; } else { s = d = e - E; }
    float v = ssrc[s] + sdst[d];
    v = (v > 0.0f) ? v : NEG_SLOPE * v;
    e_buf[e] = v;
    atomicMax(&m_enc[d], enc_f32(v));
}

// ---------------------------------------------------------------------------
// Edge pass 2: ex = exp(e - m[dst]); segment-sum into denom; e_buf <- ex.
// ---------------------------------------------------------------------------
__global__ void k_edge_exp(const int* __restrict__ dst,
                           float* __restrict__ e_buf,
                           const unsigned* __restrict__ m_enc,
                           float* __restrict__ denom, int E, int Ep) {
    int e = blockIdx.x * blockDim.x + threadIdx.x;
    if (e >= Ep) return;
    int d = (e < E) ? dst[e] : (e - E);
    float ex = __expf(e_buf[e] - dec_f32(m_enc[d]));
    e_buf[e] = ex;
    atomicAdd(&denom[d], ex);
}

// ---------------------------------------------------------------------------
// Edge pass 3: out[dst] += hW[src] * (ex / denom[dst]).
// 16 lanes per edge, float4 per lane (64 features).
// ---------------------------------------------------------------------------
__global__ void k_edge_agg(const int* __restrict__ src, const int* __restrict__ dst,
                           const float* __restrict__ hW, const float* __restrict__ e_buf,
                           const float* __restrict__ denom, float* __restrict__ out,
                           int E, int Ep) {
    int tid = blockIdx.x * blockDim.x + threadIdx.x;
    int e = tid >> 4;
    if (e >= Ep) return;
    int fc = (tid & 15) * 4;
    int s, d;
    if (e < E) { s = src[e]; d = dst[e]; } else { s = d = e - E; }
    float alpha = e_buf[e] / denom[d];
    float4 hv = *(const float4*)(hW + (size_t)s * HIDDEN + fc);
    float* o = out + (size_t)d * HIDDEN + fc;
    atomicAdd(o + 0, hv.x * alpha);
    atomicAdd(o + 1, hv.y * alpha);
    atomicAdd(o + 2, hv.z * alpha);
    atomicAdd(o + 3, hv.w * alpha);
}

// ---------------------------------------------------------------------------
// In-place bias add (+ ReLU for layers 0,1).
// ---------------------------------------------------------------------------
__global__ void k_node_finish(float* __restrict__ hout, const float* __restrict__ bias,
                              int do_relu, int N) {
    int idx = blockIdx.x * blockDim.x + threadIdx.x;
    if (idx >= N * HIDDEN) return;
    float v = hout[idx] + bias[idx & 63];
    if (do_relu) v = fmaxf(v, 0.0f);
    hout[idx] = v;
}

// ---------------------------------------------------------------------------
// Pooling: sums[g] += h[i]; counts[g] += 1
// ---------------------------------------------------------------------------
__global__ void k_pool_init(float* __restrict__ sums, float* __restrict__ counts, int G) {
    int idx = blockIdx.x * blockDim.x + threadIdx.x;
    if (idx < G * HIDDEN) sums[idx] = 0.0f;
    if (idx < G) counts[idx] = 0.0f;
}

__global__ void k_pool(const float* __restrict__ h, const int* __restrict__ batch,
                       float* __restrict__ sums, float* __restrict__ counts, int N) {
    int tid = blockIdx.x * blockDim.x + threadIdx.x;
    int node = tid >> 4;
    if (node >= N) return;
    int fc = (tid & 15) * 4;
    int g = batch[node];
    float4 v = *(const float4*)(h + (size_t)node * HIDDEN + fc);
    float* s = sums + (size_t)g * HIDDEN + fc;
    atomicAdd(s + 0, v.x);
    atomicAdd(s + 1, v.y);
    atomicAdd(s + 2, v.z);
    atomicAdd(s + 3, v.w);
    if ((tid & 15) == 0) atomicAdd(&counts[g], 1.0f);
}

// ---------------------------------------------------------------------------
// out[g,c] = (sums[g]/max(count,1)) . W_out[:,c] + b_out[c]
// ---------------------------------------------------------------------------
__global__ void k_final(const float* __restrict__ sums, const float* __restrict__ counts,
                        const float* __restrict__ W_out, const float* __restrict__ b_out,
                        float* __restrict__ out, int G) {
    int t = blockIdx.x * blockDim.x + threadIdx.x;
    if (t >= G * 2) return;
    int g = t >> 1, c = t & 1;
    float cnt = fmaxf(counts[g], 1.0f);
    float acc = 0.0f;
    #pragma unroll
    for (int f = 0; f < HIDDEN; ++f)
        acc += (sums[g * HIDDEN + f] / cnt) * W_out[f * 2 + c];
    out[t] = acc + b_out[c];
}

// ---------------------------------------------------------------------------

extern "C" void kernel_launch(void* const* d_in, const int* in_sizes, int n_in,
                              void* d_out, int out_size, void* d_ws, size_t ws_size,
                              hipStream_t stream) {
    const int*   x      = (const int*)d_in[0];
    const int*   eidx   = (const int*)d_in[1];
    const int*   batch  = (const int*)d_in[2];
    const float* embed  = (const float*)d_in[3];
    const float* Ws     = (const float*)d_in[4];
    const float* a_srcs = (const float*)d_in[5];
    const float* a_dsts = (const float*)d_in[6];
    const float* biases = (const float*)d_in[7];
    const float* W_out  = (const float*)d_in[8];
    const float* b_out  = (const float*)d_in[9];

    const int N  = in_sizes[0];
    const int E  = in_sizes[1] / 2;
    const int Ep = E + N;
    const int G  = out_size / 2;
    const int NL = in_sizes[4] / (HIDDEN * HIDDEN);   // 3 layers

    const int* srcp = eidx;
    const int* dstp = eidx + E;

    // Workspace carve-out (all float-sized sub-buffers, packed with no padding)
    float* ws = (float*)d_ws;
    size_t off = 0;
    float*    bufA  = ws + off; off += (size_t)N * HIDDEN;
    float*    bufB  = ws + off; off += (size_t)N * HIDDEN;
    float*    hW    = ws + off; off += (size_t)N * HIDDEN;
    float*    e_buf = ws + off; off += (size_t)Ep;
    unsigned* m_enc = (unsigned*)(ws + off); off += (size_t)N;
    float*    denom = ws + off; off += (size_t)N;
    float*    ssrc  = ws + off; off += (size_t)N;
    float*    sdst  = ws + off; off += (size_t)N;
    float*    sums  = ws + off; off += (size_t)G * HIDDEN;
    float*    cnts  = ws + off; off += (size_t)G;

    const int T = 256;
    const int gridNH   = (N * HIDDEN + T - 1) / T;
    const int gridN    = (N + T - 1) / T;
    const int gridGemm = (N + 127) / 128;            // 8 waves * 16 rows / block
    const int gridEp   = (Ep + T - 1) / T;
    const int gridEp16 = (Ep * 16 + T - 1) / T;
    const int gridN16  = (N * 16 + T - 1) / T;
    const int gridGH   = (G * HIDDEN + T - 1) / T;

    k_embed<<<gridNH, T, 0, stream>>>(x, embed, bufA, N);

    float* hin  = bufA;
    float* hout = bufB;
    for (int l = 0; l < NL; ++l) {
        k_layer_init<<<gridNH, T, 0, stream>>>(hout, m_enc, denom, N);
        k_gemm_hW<<<gridGemm, T, 0, stream>>>(hin, Ws + (size_t)l * HIDDEN * HIDDEN, hW, N);
        k_node_scores<<<gridN, T, 0, stream>>>(hW, a_srcs + l * HIDDEN, a_dsts + l * HIDDEN,
                                               ssrc, sdst, N);
        k_edge_score<<<gridEp, T, 0, stream>>>(srcp, dstp, ssrc, sdst, e_buf, m_enc, E, Ep);
        k_edge_exp<<<gridEp, T, 0, stream>>>(dstp, e_buf, m_enc, denom, E, Ep);
        k_edge_agg<<<gridEp16, T, 0, stream>>>(srcp, dstp, hW, e_buf, denom, hout, E, Ep);
        k_node_finish<<<gridNH, T, 0, stream>>>(hout, biases + l * HIDDEN,
                                                (l < NL - 1) ? 1 : 0, N);
        float* tmp = hin; hin = hout; hout = tmp;
    }

    k_pool_init<<<gridGH, T, 0, stream>>>(sums, cnts, G);
    k_pool<<<gridN16, T, 0, stream>>>(hin, batch, sums, cnts, N);
    k_final<<<(G * 2 + T - 1) / T, T, 0, stream>>>(sums, cnts, W_out, b_out,
                                                   (float*)d_out, G);
}